// MobiusGRU_c_38190849196686
// MI455X (gfx1250) — compile-verified
//
#include <hip/hip_runtime.h>
#include <hip/hip_bf16.h>
#include <math.h>

// ---------------------------------------------------------------------------
// Hyperbolic (Mobius) GRU forward for MI455X / gfx1250 (wave32, WMMA f32).
//
//  * expmap0 of chunked input  -> ws.inp0                     (parallel)
//  * weights packed into WMMA B-fragment order -> ws.P*       (parallel)
//  * Ux = mobius_add_chunk(U,x) for 3 gates, all T*B rows
//        via V_WMMA_F32_16X16X4_F32, packed-B inner loop      (parallel)
//  * sequential scan per layer: single workgroup, h + temps in LDS
//        (4 x 32x272 f32 ~ 140KB; CDNA5 has 320KB LDS/WGP),
//        3 recurrent 32x257x257 WMMA GEMMs per step, packed B from L2.
//  * final logmap0 of last hidden states -> ht tail of d_out.
// ---------------------------------------------------------------------------

typedef __attribute__((ext_vector_type(2))) float v2f;
typedef __attribute__((ext_vector_type(8))) float v8f;

constexpr int Tc   = 1024;
constexpr int Bc   = 32;
constexpr int INc  = 256;
constexpr int Hc   = 256;
constexpr int HPc  = 257;   // H + 1
constexpr int INPc = 260;   // NB * (IN/NB + 1)
constexpr int TBc  = Tc * Bc;
constexpr int LDC  = 272;   // padded row length (17 * 16)

// packed-weight geometry
constexpr int KS_HH = 65;                         // k-steps for 257-K GEMM (pad 260)
constexpr int KS_IH = 17;                         // k-steps per 65-chunk (pad 68)
constexpr int PWHH_FLOATS = 3 * 17 * KS_HH * 64;  // 212160
constexpr int PWIH_FLOATS = 3 * 4 * 17 * KS_IH * 64;  // 221952

#define EPS_F  1e-15f
#define CLIPF  0.99999f

__device__ __forceinline__ float wave_sum(float v) {
#pragma unroll
  for (int off = 16; off; off >>= 1) v += __shfl_xor(v, off, 32);
  return v;
}

__device__ __forceinline__ float tan_k_f(float x, float s)   { return tanhf(x * s) / s; }
__device__ __forceinline__ float artan_k_f(float x, float s) {
  float v = fminf(fmaxf(x * s, -CLIPF), CLIPF);
  return atanhf(v) / s;
}
__device__ __forceinline__ float sigmoid_f(float x) { return 1.f / (1.f + expf(-x)); }

__device__ __forceinline__ v8f wmma_f32x4(v2f a, v2f b, v8f c) {
  // D = A(16x4 f32) * B(4x16 f32) + C(16x16 f32)
  return __builtin_amdgcn_wmma_f32_16x16x4_f32(false, a, false, b, (short)0, c,
                                               false, false);
}

// ---------------------------------------------------------------------------
// Pack w_hh into B-fragment order:
//   P[((g*17+nt)*65+ks)*32 + lane] = { W[n][kb], W[n][kb+1] }   (v2f)
//   n = nt*16 + (lane&15); kb = ks*4 + 2*(lane>>4); zero pad past 257.
// ---------------------------------------------------------------------------
__global__ __launch_bounds__(256) void pack_whh_kernel(const float* __restrict__ W,
                                                       float* __restrict__ P) {
  int idx = blockIdx.x * 256 + threadIdx.x;
  const int total = 3 * 17 * KS_HH * 32;
  if (idx >= total) return;
  int l  = idx & 31;
  int ks = (idx >> 5) % KS_HH;
  int nt = ((idx >> 5) / KS_HH) % 17;
  int g  = (idx >> 5) / (KS_HH * 17);
  int n  = nt * 16 + (l & 15);
  int kb = ks * 4 + ((l >> 4) << 1);
  float v0 = 0.f, v1 = 0.f;
  if (n < HPc) {
    const size_t base = (size_t)(g * HPc + n) * HPc;
    if (kb < HPc)     v0 = W[base + kb];
    if (kb + 1 < HPc) v1 = W[base + kb + 1];
  }
  P[(size_t)idx * 2]     = v0;
  P[(size_t)idx * 2 + 1] = v1;
}

// ---------------------------------------------------------------------------
// Pack w_ih (chunked) into B-fragment order:
//   P[(((g*4+c)*17+nt)*17+ks)*32 + lane] = { W[n][coff+kb], W[n][coff+kb+1] }
// ---------------------------------------------------------------------------
__global__ __launch_bounds__(256) void pack_wih_kernel(const float* __restrict__ W,
                                                       int ldx, int cs3,
                                                       float* __restrict__ P) {
  int idx = blockIdx.x * 256 + threadIdx.x;
  const int total = 3 * 4 * 17 * KS_IH * 32;
  if (idx >= total) return;
  int l   = idx & 31;
  int ks  = (idx >> 5) % KS_IH;
  int rem = (idx >> 5) / KS_IH;
  int nt  = rem % 17; rem /= 17;
  int c   = rem % 4;
  int g   = rem / 4;
  int cs   = (c == 3) ? cs3 : 65;
  int coff = c * 65;
  int n  = nt * 16 + (l & 15);
  int kb = ks * 4 + ((l >> 4) << 1);
  float v0 = 0.f, v1 = 0.f;
  if (n < HPc) {
    const size_t base = (size_t)(g * HPc + n) * ldx + coff;
    if (kb < cs)     v0 = W[base + kb];
    if (kb + 1 < cs) v1 = W[base + kb + 1];
  }
  P[(size_t)idx * 2]     = v0;
  P[(size_t)idx * 2 + 1] = v1;
}

// ---------------------------------------------------------------------------
// embed: xb -> prepend 0 per 64-chunk, expmap0 over 65-dim chunk -> inp0
// one wave per (t,b,chunk)
// ---------------------------------------------------------------------------
__global__ __launch_bounds__(256) void embed_kernel(const float* __restrict__ X,
                                                    float* __restrict__ inp0,
                                                    const float* __restrict__ kp) {
  const float kv = kp[0];
  const float s  = sqrtf(-kv);
  int wid  = blockIdx.x * 8 + (threadIdx.x >> 5);   // 0 .. T*B*4-1
  int lane = threadIdx.x & 31;
  int c  = wid & 3;
  int tb = wid >> 2;
  const float* xr = X + (size_t)tb * INc + c * 64;
  float v0 = xr[lane], v1 = xr[lane + 32];
  float ss = wave_sum(v0 * v0 + v1 * v1);
  float n  = sqrtf(fmaxf(ss, EPS_F));
  float f  = tan_k_f(n, s) / n;
  float* o = inp0 + (size_t)tb * INPc + c * 65;
  if (lane == 0) o[0] = 0.f;
  o[1 + lane]  = f * v0;
  o[33 + lane] = f * v1;
}

// ---------------------------------------------------------------------------
// bias expmap0: bh[2][3][257]
// ---------------------------------------------------------------------------
__global__ __launch_bounds__(256) void bias_kernel(const float* __restrict__ b0,
                                                   const float* __restrict__ b1,
                                                   float* __restrict__ bh,
                                                   const float* __restrict__ kp) {
  const float kv = kp[0];
  const float s  = sqrtf(-kv);
  int lane = threadIdx.x & 31, wv = threadIdx.x >> 5;
  if (wv >= 6) return;
  int l = wv / 3, r = wv % 3;
  const float* src = (l ? b1 : b0) + (size_t)r * HPc;
  float ss = 0.f;
  for (int j = lane; j < HPc; j += 32) { float v = src[j]; ss += v * v; }
  ss = wave_sum(ss);
  float n = sqrtf(fmaxf(ss, EPS_F));
  float f = tan_k_f(n, s) / n;
  float* dst = bh + ((size_t)l * 3 + r) * HPc;
  for (int j = lane; j < HPc; j += 32) dst[j] = f * src[j];
}

// ---------------------------------------------------------------------------
// Ux precompute: mobius_add_chunk(U, x) for 3 gates, 16 rows per block.
// B read from packed Pwih -> guard-free WMMA inner loop.
// ---------------------------------------------------------------------------
__global__ __launch_bounds__(256) void ux_kernel(const float* __restrict__ X, int ldx,
                                                 int cs3,
                                                 const float* __restrict__ Pwih,
                                                 float* __restrict__ Ux,
                                                 const float* __restrict__ kp) {
  __shared__ float xsh[16][68];
  __shared__ float mxs[16][LDC];
  __shared__ float acc[3][16][LDC];
  __shared__ float xn2[16], sc[16], a2[3][16], rA[16], rB[16];

  const float kv = kp[0];
  const float s  = sqrtf(-kv);
  const int m0   = blockIdx.x * 16;
  const int tid  = threadIdx.x, lane = tid & 31, wv = tid >> 5;
  const int half = lane >> 4, rrow = lane & 15;
  const v2f* Pw = (const v2f*)Pwih;

  for (int c = 0; c < 4; ++c) {
    const int coff = c * 65;
    const int cs   = (c == 3) ? cs3 : 65;
    // stage x chunk (zero padded to 68)
    for (int i = tid; i < 16 * 68; i += 256) {
      int r = i / 68, j = i % 68;
      xsh[r][j] = (j < cs) ? X[(size_t)(m0 + r) * ldx + coff + j] : 0.f;
    }
    __syncthreads();
    // per-row ||x_c||^2
    for (int rr = 0; rr < 2; ++rr) {
      int row = wv + rr * 8;
      float ss = 0.f;
      for (int j = lane; j < cs; j += 32) { float v = xsh[row][j]; ss += v * v; }
      ss = wave_sum(ss);
      if (!lane) xn2[row] = ss;
    }
    __syncthreads();

    for (int g = 0; g < 3; ++g) {
      // mx = x_c @ U_{g,c}^T via WMMA f32 16x16x4, guard-free packed B
      for (int nt = wv; nt < 17; nt += 8) {
        const v2f* bp = Pw + ((size_t)((g * 4 + c) * 17 + nt) * KS_IH) * 32 + lane;
        v8f cacc = {0, 0, 0, 0, 0, 0, 0, 0};
#pragma unroll
        for (int ks = 0; ks < KS_IH; ++ks) {
          const int kk = ks * 4 + (half << 1);
          v2f a = *reinterpret_cast<const v2f*>(&xsh[rrow][kk]);
          v2f b = bp[(size_t)ks * 32];
          cacc = wmma_f32x4(a, b, cacc);
        }
#pragma unroll
        for (int r = 0; r < 8; ++r) mxs[(half << 3) + r][nt * 16 + rrow] = cacc[r];
      }
      __syncthreads();
      // mobius_matvec scale per row
      for (int rr = 0; rr < 2; ++rr) {
        int row = wv + rr * 8;
        float ss = 0.f;
        for (int j = lane; j < HPc; j += 32) { float v = mxs[row][j]; ss += v * v; }
        ss = wave_sum(ss);
        if (!lane) {
          float xn  = sqrtf(fmaxf(xn2[row], EPS_F));
          float mxn = sqrtf(fmaxf(ss, EPS_F));
          float scl = tan_k_f(mxn / xn * artan_k_f(xn, s), s) / mxn;
          sc[row] = scl;
          rA[row] = scl * scl * ss;   // ||y||^2
        }
      }
      __syncthreads();
      if (c == 0) {
        for (int rr = 0; rr < 2; ++rr) {
          int row = wv + rr * 8;
          for (int j = lane; j < HPc; j += 32) acc[g][row][j] = sc[row] * mxs[row][j];
        }
        if (tid < 16) a2[g][tid] = rA[tid];
        __syncthreads();
      } else {
        // mobius_add(acc, y)
        for (int rr = 0; rr < 2; ++rr) {
          int row = wv + rr * 8;
          float sxy = 0.f;
          for (int j = lane; j < HPc; j += 32) sxy += acc[g][row][j] * mxs[row][j];
          sxy = wave_sum(sxy);
          if (!lane) {
            float xy = sc[row] * sxy, x2 = a2[g][row], y2 = rA[row];
            float c1  = 1.f - 2.f * kv * xy - kv * y2;
            float c2  = 1.f + kv * x2;
            float den = fmaxf(1.f - 2.f * kv * xy + kv * kv * x2 * y2, EPS_F);
            rB[row] = c1 / den;
            sc[row] = sc[row] * c2 / den;
          }
        }
        __syncthreads();
        for (int rr = 0; rr < 2; ++rr) {
          int row = wv + rr * 8;
          for (int j = lane; j < HPc; j += 32)
            acc[g][row][j] = rB[row] * acc[g][row][j] + sc[row] * mxs[row][j];
        }
        __syncthreads();
        if (c < 3) {
          for (int rr = 0; rr < 2; ++rr) {
            int row = wv + rr * 8;
            float ss = 0.f;
            for (int j = lane; j < HPc; j += 32) { float v = acc[g][row][j]; ss += v * v; }
            ss = wave_sum(ss);
            if (!lane) a2[g][row] = ss;
          }
        }
        __syncthreads();
      }
    }  // gates
  }    // chunks
  for (int g = 0; g < 3; ++g)
    for (int rr = 0; rr < 2; ++rr) {
      int row = wv + rr * 8;
      for (int j = lane; j < HPc; j += 32)
        Ux[((size_t)g * TBc + m0 + row) * HPc + j] = acc[g][row][j];
    }
}

// ---------------------------------------------------------------------------
// Recurrent GEMM: C(32x257) = A(32x257) @ W_g^T, WMMA f32, packed B.
// A is LDS row-major [32][LDC]; cols 257..259 must be zero.
// ---------------------------------------------------------------------------
__device__ __forceinline__ void block_gemm_257(const float (*A)[LDC],
                                               const v2f* __restrict__ P, int g,
                                               float (*C)[LDC], int tid) {
  const int lane = tid & 31, wv = tid >> 5;
  const int half = lane >> 4, rrow = lane & 15;
  for (int ti = wv; ti < 34; ti += 8) {
    const int mt = ti & 1, nt = ti >> 1;
    const v2f* bp = P + ((size_t)(g * 17 + nt) * KS_HH) * 32 + lane;
    v8f cacc = {0, 0, 0, 0, 0, 0, 0, 0};
#pragma unroll 5
    for (int ks = 0; ks < KS_HH; ++ks) {
      const int kk = ks * 4 + (half << 1);
      v2f a = *reinterpret_cast<const v2f*>(&A[mt * 16 + rrow][kk]);
      v2f b = bp[(size_t)ks * 32];
      cacc = wmma_f32x4(a, b, cacc);
    }
#pragma unroll
    for (int r = 0; r < 8; ++r)
      C[mt * 16 + (half << 3) + r][nt * 16 + rrow] = cacc[r];
  }
}

// one_rnn_transform without the final logmap:
//   C = mobius_add(mobius_add(mobius_matvec(W_g, A), Ux_g), b_g)
__device__ void gate_pre(const float (*A)[LDC], const float* an2s,
                         const v2f* __restrict__ P, int g,
                         const float* __restrict__ ux,
                         const float* __restrict__ brow,
                         float (*C)[LDC], float* rA, float* rB,
                         float kv, float s, int tid) {
  const int lane = tid & 31, wv = tid >> 5;
  block_gemm_257(A, P, g, C, tid);
  __syncthreads();
  // mobius_matvec scaling
  for (int rr = 0; rr < 4; ++rr) {
    int row = wv + rr * 8;
    float ss = 0.f;
    for (int j = lane; j < HPc; j += 32) { float v = C[row][j]; ss += v * v; }
    ss = wave_sum(ss);
    if (!lane) {
      float xn  = sqrtf(fmaxf(an2s[row], EPS_F));
      float mxn = sqrtf(fmaxf(ss, EPS_F));
      float scl = tan_k_f(mxn / xn * artan_k_f(xn, s), s) / mxn;
      rA[row] = scl;
      rB[row] = scl * scl * ss;           // ||Wh||^2 after scaling
    }
  }
  __syncthreads();
  for (int rr = 0; rr < 4; ++rr) {
    int row = wv + rr * 8;
    for (int j = lane; j < HPc; j += 32) C[row][j] *= rA[row];
  }
  __syncthreads();
  // mobius_add with Ux (from L2)
  for (int rr = 0; rr < 4; ++rr) {
    int row = wv + rr * 8;
    const float* u = ux + (size_t)row * HPc;
    float sy = 0.f, sxy = 0.f;
    for (int j = lane; j < HPc; j += 32) {
      float uvv = u[j];
      sy  += uvv * uvv;
      sxy += C[row][j] * uvv;
    }
    sy = wave_sum(sy); sxy = wave_sum(sxy);
    if (!lane) {
      float x2 = rB[row], y2 = sy, xy = sxy;
      float c1  = 1.f - 2.f * kv * xy - kv * y2;
      float c2  = 1.f + kv * x2;
      float den = fmaxf(1.f - 2.f * kv * xy + kv * kv * x2 * y2, EPS_F);
      rA[row] = c1 / den; rB[row] = c2 / den;
    }
  }
  __syncthreads();
  for (int rr = 0; rr < 4; ++rr) {
    int row = wv + rr * 8;
    const float* u = ux + (size_t)row * HPc;
    for (int j = lane; j < HPc; j += 32)
      C[row][j] = rA[row] * C[row][j] + rB[row] * u[j];
  }
  __syncthreads();
  // mobius_add with bias
  for (int rr = 0; rr < 4; ++rr) {
    int row = wv + rr * 8;
    float sx = 0.f, sxy = 0.f, sy = 0.f;
    for (int j = lane; j < HPc; j += 32) {
      float cv = C[row][j], bv = brow[j];
      sx += cv * cv; sxy += cv * bv; sy += bv * bv;
    }
    sx = wave_sum(sx); sxy = wave_sum(sxy); sy = wave_sum(sy);
    if (!lane) {
      float c1  = 1.f - 2.f * kv * sxy - kv * sy;
      float c2  = 1.f + kv * sx;
      float den = fmaxf(1.f - 2.f * kv * sxy + kv * kv * sx * sy, EPS_F);
      rA[row] = c1 / den; rB[row] = c2 / den;
    }
  }
  __syncthreads();
  for (int rr = 0; rr < 4; ++rr) {
    int row = wv + rr * 8;
    for (int j = lane; j < HPc; j += 32)
      C[row][j] = rA[row] * C[row][j] + rB[row] * brow[j];
  }
  __syncthreads();
}

// ---------------------------------------------------------------------------
// Sequential scan over T for one layer. Single workgroup, 8 waves.
// ---------------------------------------------------------------------------
__global__ __launch_bounds__(256) void scan_kernel(const float* __restrict__ Ux,
                                                   const float* __restrict__ Pwhh,
                                                   const float* __restrict__ bh,
                                                   float* __restrict__ outp,
                                                   float* __restrict__ hlast,
                                                   const float* __restrict__ kp) {
  __shared__ float h [32][LDC];
  __shared__ float t1[32][LDC];   // z
  __shared__ float t2[32][LDC];   // rh / h_int
  __shared__ float t3[32][LDC];   // gate pre-act / delta
  __shared__ float hn2[32], an2[32], rA[32], rB[32];

  const float kv = kp[0];
  const float s  = sqrtf(-kv);
  const int tid = threadIdx.x, lane = tid & 31, wv = tid >> 5;
  const v2f* P = (const v2f*)Pwhh;

  for (int i = tid; i < 32 * LDC; i += 256) {
    int r = i / LDC, j = i % LDC;
    h[r][j] = 0.f; t1[r][j] = 0.f; t2[r][j] = 0.f; t3[r][j] = 0.f;
  }
  __syncthreads();

  for (int t = 0; t < Tc; ++t) {
    const float* ux_r = Ux + ((size_t)0 * TBc + (size_t)t * Bc) * HPc;
    const float* ux_h = Ux + ((size_t)1 * TBc + (size_t)t * Bc) * HPc;
    const float* ux_z = Ux + ((size_t)2 * TBc + (size_t)t * Bc) * HPc;

    // ||h||^2 per row
    for (int rr = 0; rr < 4; ++rr) {
      int row = wv + rr * 8; float ss = 0.f;
      for (int j = lane; j < HPc; j += 32) { float v = h[row][j]; ss += v * v; }
      ss = wave_sum(ss);
      if (!lane) hn2[row] = ss;
    }
    __syncthreads();

    // z gate (g=2: W_hz rows [2*HP, 3*HP))
    gate_pre(h, hn2, P, 2, ux_z, bh + 2 * HPc, t3, rA, rB, kv, s, tid);
    for (int rr = 0; rr < 4; ++rr) {
      int row = wv + rr * 8; float ss = 0.f;
      for (int j = lane; j < HPc; j += 32) { float v = t3[row][j]; ss += v * v; }
      ss = wave_sum(ss);
      if (!lane) { float n = sqrtf(fmaxf(ss, EPS_F)); rA[row] = artan_k_f(n, s) / n; }
    }
    __syncthreads();
    for (int rr = 0; rr < 4; ++rr) {
      int row = wv + rr * 8;
      for (int j = lane; j < HPc; j += 32)
        t1[row][j] = sigmoid_f(rA[row] * t3[row][j]);
    }
    __syncthreads();

    // r gate (g=0: W_hr rows [0, HP))
    gate_pre(h, hn2, P, 0, ux_r, bh, t3, rA, rB, kv, s, tid);
    for (int rr = 0; rr < 4; ++rr) {
      int row = wv + rr * 8; float ss = 0.f;
      for (int j = lane; j < HPc; j += 32) { float v = t3[row][j]; ss += v * v; }
      ss = wave_sum(ss);
      if (!lane) { float n = sqrtf(fmaxf(ss, EPS_F)); rA[row] = artan_k_f(n, s) / n; }
    }
    __syncthreads();
    for (int rr = 0; rr < 4; ++rr) {
      int row = wv + rr * 8;
      for (int j = lane; j < HPc; j += 32)
        t3[row][j] = sigmoid_f(rA[row] * t3[row][j]);   // r in place
    }
    __syncthreads();

    // rh = expmap0(r * logmap0(h)) -> t2 ; an2 = ||rh||^2
    for (int rr = 0; rr < 4; ++rr) {
      int row = wv + rr * 8;
      float hn = sqrtf(fmaxf(hn2[row], EPS_F));
      float ch = artan_k_f(hn, s) / hn;
      float ss = 0.f;
      for (int j = lane; j < HPc; j += 32) {
        float v = t3[row][j] * ch * h[row][j];
        ss += v * v;
      }
      ss = wave_sum(ss);
      if (!lane) {
        float nv = sqrtf(fmaxf(ss, EPS_F));
        float f  = tan_k_f(nv, s) / nv;
        rA[row]  = f * ch;
        an2[row] = f * f * ss;
      }
    }
    __syncthreads();
    for (int rr = 0; rr < 4; ++rr) {
      int row = wv + rr * 8;
      for (int j = lane; j < HPc; j += 32)
        t2[row][j] = rA[row] * t3[row][j] * h[row][j];
    }
    __syncthreads();

    // h_tilde -> t3 (g=1: W_hhh rows [HP, 2*HP))
    gate_pre(t2, an2, P, 1, ux_h, bh + HPc, t3, rA, rB, kv, s, tid);

    // delta = mobius_add(proj(-h), h_tilde) -> t3 (in place)
    for (int rr = 0; rr < 4; ++rr) {
      int row = wv + rr * 8;
      float sy = 0.f, sxy = 0.f;
      for (int j = lane; j < HPc; j += 32) {
        float y = t3[row][j];
        sy += y * y; sxy += h[row][j] * y;
      }
      sy = wave_sum(sy); sxy = wave_sum(sxy);
      if (!lane) {
        float n     = sqrtf(fmaxf(hn2[row], EPS_F));
        float maxn  = CLIPF / s;
        float alpha = (n > maxn) ? (maxn / n) : 1.f;
        float x2 = alpha * alpha * hn2[row];
        float xy = -alpha * sxy, y2 = sy;
        float c1  = 1.f - 2.f * kv * xy - kv * y2;
        float c2  = 1.f + kv * x2;
        float den = fmaxf(1.f - 2.f * kv * xy + kv * kv * x2 * y2, EPS_F);
        rA[row] = -alpha * c1 / den;
        rB[row] = c2 / den;
      }
    }
    __syncthreads();
    for (int rr = 0; rr < 4; ++rr) {
      int row = wv + rr * 8;
      for (int j = lane; j < HPc; j += 32)
        t3[row][j] = rA[row] * h[row][j] + rB[row] * t3[row][j];
    }
    __syncthreads();

    // h_int = ptransp0(h, z * logmap0(delta)) -> t2
    for (int rr = 0; rr < 4; ++rr) {
      int row = wv + rr * 8; float ss = 0.f;
      for (int j = lane; j < HPc; j += 32) { float v = t3[row][j]; ss += v * v; }
      ss = wave_sum(ss);
      if (!lane) {
        float nd = sqrtf(fmaxf(ss, EPS_F));
        float cd = artan_k_f(nd, s) / nd;
        float pt = fmaxf(1.f + kv * hn2[row], EPS_F);
        rA[row] = cd * pt;
      }
    }
    __syncthreads();
    for (int rr = 0; rr < 4; ++rr) {
      int row = wv + rr * 8;
      for (int j = lane; j < HPc; j += 32)
        t2[row][j] = t1[row][j] * rA[row] * t3[row][j];
    }
    __syncthreads();

    // h_out = expmap(h_int, h) -> h ; also emit out[t]
    for (int rr = 0; rr < 4; ++rr) {
      int row = wv + rr * 8;
      float sv = 0.f, shv = 0.f;
      for (int j = lane; j < HPc; j += 32) {
        float v = t2[row][j];
        sv += v * v; shv += h[row][j] * v;
      }
      sv = wave_sum(sv); shv = wave_sum(shv);
      if (!lane) {
        float lam  = 2.f / fmaxf(1.f + kv * hn2[row], EPS_F);
        float n    = sqrtf(fmaxf(sv, EPS_F));
        float beta = tan_k_f(0.5f * lam * n, s) / n;
        float w2 = beta * beta * sv, xy = beta * shv, x2 = hn2[row];
        float c1  = 1.f - 2.f * kv * xy - kv * w2;
        float c2  = 1.f + kv * x2;
        float den = fmaxf(1.f - 2.f * kv * xy + kv * kv * x2 * w2, EPS_F);
        rA[row] = c1 / den;
        rB[row] = beta * c2 / den;
      }
    }
    __syncthreads();
    for (int rr = 0; rr < 4; ++rr) {
      int row = wv + rr * 8;
      float* orow = outp + ((size_t)t * Bc + row) * HPc;
      for (int j = lane; j < HPc; j += 32) {
        float hv = rA[row] * h[row][j] + rB[row] * t2[row][j];
        h[row][j] = hv;
        orow[j] = hv;
      }
    }
    __syncthreads();
  }

  for (int rr = 0; rr < 4; ++rr) {
    int row = wv + rr * 8;
    for (int j = lane; j < HPc; j += 32) hlast[(size_t)row * HPc + j] = h[row][j];
  }
}

// ---------------------------------------------------------------------------
// ht = stack(logmap0(hl0)[:,1:], logmap0(hl1)[:,1:]) -> tail of d_out
// ---------------------------------------------------------------------------
__global__ __launch_bounds__(256) void final_kernel(const float* __restrict__ hlast,
                                                    float* __restrict__ out,
                                                    const float* __restrict__ kp) {
  const float kv = kp[0];
  const float s  = sqrtf(-kv);
  const int lane = threadIdx.x & 31, wv = threadIdx.x >> 5;
  for (int row = wv; row < 64; row += 8) {
    const int l = row >> 5, b = row & 31;
    const float* src = hlast + (size_t)row * HPc;
    float ss = 0.f;
    for (int j = lane; j < HPc; j += 32) { float v = src[j]; ss += v * v; }
    ss = wave_sum(ss);
    float n = sqrtf(fmaxf(ss, EPS_F));
    float f = artan_k_f(n, s) / n;
    float* dst = out + (size_t)Tc * Bc * HPc + ((size_t)l * Bc + b) * Hc;
    for (int j = lane; j < Hc; j += 32) dst[j] = f * src[1 + j];
  }
}

// ---------------------------------------------------------------------------
extern "C" void kernel_launch(void* const* d_in, const int* in_sizes, int n_in,
                              void* d_out, int out_size, void* d_ws, size_t ws_size,
                              hipStream_t stream) {
  (void)in_sizes; (void)n_in; (void)out_size; (void)ws_size;
  const float* x    = (const float*)d_in[0];
  const float* kp   = (const float*)d_in[1];
  const float* wih0 = (const float*)d_in[2];
  const float* whh0 = (const float*)d_in[3];
  const float* b0   = (const float*)d_in[4];
  const float* wih1 = (const float*)d_in[5];
  const float* whh1 = (const float*)d_in[6];
  const float* b1   = (const float*)d_in[7];
  float* out = (float*)d_out;

  float* ws    = (float*)d_ws;
  float* inp0  = ws;                                // TB * INP
  float* out0  = inp0 + (size_t)TBc * INPc;         // TB * HP
  float* Ux    = out0 + (size_t)TBc * HPc;          // 3 * TB * HP
  float* bh    = Ux + (size_t)3 * TBc * HPc;        // 2 * 3 * HP
  float* hlast = bh + (size_t)2 * 3 * HPc;          // 2 * 32 * HP
  float* Pwih  = hlast + (size_t)2 * 32 * HPc;      // PWIH_FLOATS
  float* Pwhh  = Pwih + (size_t)PWIH_FLOATS;        // PWHH_FLOATS

  const int packIhBlocks = (3 * 4 * 17 * KS_IH * 32 + 255) / 256;
  const int packHhBlocks = (3 * 17 * KS_HH * 32 + 255) / 256;

  embed_kernel<<<TBc * 4 / 8, 256, 0, stream>>>(x, inp0, kp);
  bias_kernel<<<1, 256, 0, stream>>>(b0, b1, bh, kp);

  // layer 0
  pack_wih_kernel<<<packIhBlocks, 256, 0, stream>>>(wih0, INPc, 65, Pwih);
  ux_kernel<<<TBc / 16, 256, 0, stream>>>(inp0, INPc, 65, Pwih, Ux, kp);
  pack_whh_kernel<<<packHhBlocks, 256, 0, stream>>>(whh0, Pwhh);
  scan_kernel<<<1, 256, 0, stream>>>(Ux, Pwhh, bh, out0, hlast, kp);

  // layer 1 (input dim 257, last chunk size 62)
  pack_wih_kernel<<<packIhBlocks, 256, 0, stream>>>(wih1, HPc, 62, Pwih);
  ux_kernel<<<TBc / 16, 256, 0, stream>>>(out0, HPc, 62, Pwih, Ux, kp);
  pack_whh_kernel<<<packHhBlocks, 256, 0, stream>>>(whh1, Pwhh);
  scan_kernel<<<1, 256, 0, stream>>>(Ux, Pwhh, bh + 3 * HPc, out, hlast + 32 * HPc,
                                     kp);

  final_kernel<<<1, 256, 0, stream>>>(hlast, out, kp);
}